// BiLSTM_CRF_Char_8641474200065
// MI455X (gfx1250) — compile-verified
//
#include <hip/hip_runtime.h>
#include <math.h>

// ---------------------------------------------------------------------------
// BiLSTM-CRF tagger for gfx1250 (MI455X), fp32 end-to-end using
// V_WMMA_F32_16X16X4_F32 for every GEMM in the LSTM recurrences.
// Word LSTM is split 8-ways across the hidden dimension per direction with a
// per-step device barrier so the sequential recurrence runs on 16 WGPs.
// ---------------------------------------------------------------------------

#define S_LEN 128
#define B_SZ  32
#define CLEN  16
#define DW    256
#define DC    64
#define HW2   256   // word hidden per direction
#define HC2   64    // char hidden per direction
#define NTAG  17
#define BS    (B_SZ * S_LEN)   // 4096
#define NSLICE 8               // word-LSTM workgroups per direction

typedef float v2f __attribute__((ext_vector_type(2)));
typedef float v8f __attribute__((ext_vector_type(8)));

__device__ __forceinline__ v8f wmma4(v2f a, v2f b, v8f c) {
  // D = A(16x4) * B(4x16) + C(16x16), all fp32
  return __builtin_amdgcn_wmma_f32_16x16x4_f32(false, a, false, b, (short)0, c,
                                               false, false);
}

__device__ __forceinline__ float sigm(float x) { return 1.0f / (1.0f + __expf(-x)); }

// ---------------------------------------------------------------------------
// Gather kernels
// ---------------------------------------------------------------------------

// CE[t][n][d] = W_ce[char[n*16+t]][d], n = b*S+s, layout (C, B*S, DC)
__global__ void k_char_gather(const int* __restrict__ ch,
                              const float* __restrict__ Wce,
                              float* __restrict__ CE) {
  int e = blockIdx.x * blockDim.x + threadIdx.x;
  if (e >= CLEN * BS * DC) return;
  int d = e & (DC - 1);
  int n = (e >> 6) & (BS - 1);
  int t = e >> 18;
  int ci = ch[n * CLEN + t];
  CE[e] = Wce[ci * DC + d];
}

// XW[s][b][128+d] = W_we[sentence[s][b]][d]   (XW is (S,B,384))
__global__ void k_word_gather(const int* __restrict__ sent,
                              const float* __restrict__ Wwe,
                              float* __restrict__ XW) {
  int e = blockIdx.x * blockDim.x + threadIdx.x;
  if (e >= S_LEN * B_SZ * DW) return;
  int d = e & (DW - 1);
  int b = (e >> 8) & (B_SZ - 1);
  int s = e >> 13;
  int wi = sent[s * B_SZ + b];
  XW[((size_t)(s * B_SZ + b)) * 384 + 128 + d] = Wwe[(size_t)wi * DW + d];
}

// ---------------------------------------------------------------------------
// Weight transpose prep: W'[k][n] = concat(Wih, Whh)[n][k]
// ---------------------------------------------------------------------------

__global__ void k_prep_wc(const float* __restrict__ Wih_f, const float* __restrict__ Whh_f,
                          const float* __restrict__ Wih_b, const float* __restrict__ Whh_b,
                          float* __restrict__ WcT) {
  int e = blockIdx.x * blockDim.x + threadIdx.x;
  if (e >= 2 * 128 * 256) return;
  int n = e & 255;
  int k = (e >> 8) & 127;
  int dir = e >> 15;
  const float* Wih = dir ? Wih_b : Wih_f;
  const float* Whh = dir ? Whh_b : Whh_f;
  WcT[e] = (k < 64) ? Wih[n * 64 + k] : Whh[n * 64 + (k - 64)];
}

__global__ void k_prep_ww(const float* __restrict__ Wih_f, const float* __restrict__ Whh_f,
                          const float* __restrict__ Wih_b, const float* __restrict__ Whh_b,
                          float* __restrict__ WwT) {
  int e = blockIdx.x * blockDim.x + threadIdx.x;
  if (e >= 2 * 640 * 1024) return;
  int n = e & 1023;
  int kd = e >> 10;
  int k = kd % 640;
  int dir = kd / 640;
  const float* Wih = dir ? Wih_b : Wih_f;
  const float* Whh = dir ? Whh_b : Whh_f;
  WwT[e] = (k < 384) ? Wih[n * 384 + k] : Whh[n * 256 + (k - 384)];
}

// Zero the word-LSTM hidden double-buffer and the per-step barrier counters.
// Runs every launch so graph replays stay deterministic.
__global__ void k_init(float* __restrict__ HG, unsigned* __restrict__ CNT) {
  int e = blockIdx.x * blockDim.x + threadIdx.x;
  if (e < 2 * 2 * B_SZ * HW2) HG[e] = 0.0f;
  if (e < 2 * S_LEN) CNT[e] = 0u;
}

// ---------------------------------------------------------------------------
// Char LSTM: one workgroup = (32-row batch tile, direction). 16 steps.
// Per step: gates(32x256) = [x_t, h](32x128) @ W'(128x256) via WMMA.
// Writes final hidden state into XW[:, :, dir*64 + j].
// ---------------------------------------------------------------------------

__global__ void __launch_bounds__(256) k_char_lstm(
    const float* __restrict__ CE, const float* __restrict__ WcT,
    const float* __restrict__ bf, const float* __restrict__ bb,
    float* __restrict__ XW) {
  const int dir = blockIdx.y;
  const int rowbase = blockIdx.x * 32;
  const float* bias = dir ? bb : bf;
  const float* Wt = WcT + (size_t)dir * 128 * 256;

  __shared__ float sh_h[32][68];   // stride 68 -> bank step 4, conflict-free A frags
  __shared__ float sh_c[32][64];
  __shared__ float sh_g[32][260];  // stride 260

  for (int idx = threadIdx.x; idx < 32 * 64; idx += 256) {
    int m = idx >> 6, j = idx & 63;
    sh_h[m][j] = 0.0f;
    sh_c[m][j] = 0.0f;
  }
  __syncthreads();

  const int lane  = threadIdx.x & 31;
  const int wave  = threadIdx.x >> 5;    // 0..7
  const int laneM = lane & 15;
  const int laneG = lane >> 4;           // 0 or 1
  const int laneK = laneG << 1;          // 0 or 2

  for (int t = 0; t < CLEN; ++t) {
    const int tc = dir ? (CLEN - 1 - t) : t;
    const float* X = CE + ((size_t)tc * BS + rowbase) * DC;

    v8f acc[4];
    for (int i = 0; i < 4; ++i)
      for (int j = 0; j < 8; ++j) acc[i][j] = 0.0f;

    const int n0 = (wave * 2) * 16 + laneM;
    for (int k0 = 0; k0 < 128; k0 += 4) {
      const int k = k0 + laneK;
      v2f a0, a1;
      if (k0 < 64) {
        a0.x = X[laneM * DC + k];          a0.y = X[laneM * DC + k + 1];
        a1.x = X[(16 + laneM) * DC + k];   a1.y = X[(16 + laneM) * DC + k + 1];
      } else {
        a0.x = sh_h[laneM][k - 64];        a0.y = sh_h[laneM][k - 63];
        a1.x = sh_h[16 + laneM][k - 64];   a1.y = sh_h[16 + laneM][k - 63];
      }
      v2f b0, b1;
      b0.x = Wt[k * 256 + n0];        b0.y = Wt[(k + 1) * 256 + n0];
      b1.x = Wt[k * 256 + n0 + 16];   b1.y = Wt[(k + 1) * 256 + n0 + 16];
      acc[0] = wmma4(a0, b0, acc[0]);
      acc[1] = wmma4(a1, b0, acc[1]);
      acc[2] = wmma4(a0, b1, acc[2]);
      acc[3] = wmma4(a1, b1, acc[3]);
    }

    for (int i = 0; i < 4; ++i) {
      int mbase = (i & 1) * 16;
      int nbase = (wave * 2 + (i >> 1)) * 16;
      for (int j = 0; j < 8; ++j)
        sh_g[mbase + j + 8 * laneG][nbase + laneM] = acc[i][j];
    }
    __syncthreads();

    for (int idx = threadIdx.x; idx < 32 * 64; idx += 256) {
      int m = idx >> 6, j = idx & 63;
      float gi = sh_g[m][j]        + bias[j];
      float gf = sh_g[m][64 + j]   + bias[64 + j];
      float gg = sh_g[m][128 + j]  + bias[128 + j];
      float go = sh_g[m][192 + j]  + bias[192 + j];
      float c  = sigm(gf) * sh_c[m][j] + sigm(gi) * tanhf(gg);
      sh_c[m][j] = c;
      sh_h[m][j] = sigm(go) * tanhf(c);
    }
    __syncthreads();
  }

  for (int idx = threadIdx.x; idx < 32 * 64; idx += 256) {
    int m = idx >> 6, j = idx & 63;
    int n = rowbase + m;          // n = b*S + s
    int b = n >> 7, s = n & (S_LEN - 1);
    XW[((size_t)(s * B_SZ + b)) * 384 + dir * 64 + j] = sh_h[m][j];
  }
}

// ---------------------------------------------------------------------------
// Word LSTM, hidden-split: grid (NSLICE, 2). WG `slice` of direction `dir`
// owns hidden cols [slice*32, slice*32+32), i.e. gate columns
// {g*256 + j : g<4}. Full h is exchanged through a double-buffered global
// array HG with a per-step one-shot atomic barrier (counters in CNT).
// ---------------------------------------------------------------------------

__global__ void __launch_bounds__(256) k_word_lstm(
    const float* __restrict__ XW, const float* __restrict__ WwT,
    const float* __restrict__ bf, const float* __restrict__ bb,
    float* __restrict__ OUT, float* __restrict__ HG, unsigned* __restrict__ CNT) {
  const int slice = blockIdx.x;          // 0..7
  const int dir   = blockIdx.y;          // 0..1
  const int j0    = slice * 32;          // hidden-slice base
  const float* bias = dir ? bb : bf;
  const float* Wt = WwT + (size_t)dir * 640 * 1024;

  __shared__ float sh_h[32][260];  // staged full h (prev step), stride 260
  __shared__ float sh_c[32][32];   // cell state, this slice only
  __shared__ float sh_g[32][132];  // gate slice (4 gates x 32), stride 132

  for (int idx = threadIdx.x; idx < 32 * 32; idx += 256)
    sh_c[idx >> 5][idx & 31] = 0.0f;

  const int lane  = threadIdx.x & 31;
  const int wave  = threadIdx.x >> 5;    // 0..7
  const int laneM = lane & 15;
  const int laneG = lane >> 4;
  const int laneK = laneG << 1;
  // this wave's gate column: gate g = wave>>1, 16-col half = wave&1
  const int nc = (wave >> 1) * 256 + j0 + (wave & 1) * 16 + laneM;
  const int lcol = (wave >> 1) * 32 + (wave & 1) * 16 + laneM;  // col in sh_g

  for (int t = 0; t < S_LEN; ++t) {
    const int tw = dir ? (S_LEN - 1 - t) : t;
    const float* X = XW + (size_t)tw * B_SZ * 384;

    // stage previous h (buffer (t+1)&1; buffer 1 pre-zeroed for t=0)
    const float* hsrc = HG + ((size_t)(dir * 2 + ((t + 1) & 1))) * B_SZ * HW2;
    for (int idx = threadIdx.x; idx < B_SZ * HW2; idx += 256)
      sh_h[idx >> 8][idx & 255] = hsrc[idx];
    __syncthreads();

    v8f acc0, acc1;
    for (int j = 0; j < 8; ++j) { acc0[j] = 0.0f; acc1[j] = 0.0f; }

    for (int k0 = 0; k0 < 640; k0 += 4) {
      const int k = k0 + laneK;
      v2f a0, a1;
      if (k0 < 384) {
        a0.x = X[laneM * 384 + k];            a0.y = X[laneM * 384 + k + 1];
        a1.x = X[(16 + laneM) * 384 + k];     a1.y = X[(16 + laneM) * 384 + k + 1];
      } else {
        a0.x = sh_h[laneM][k - 384];          a0.y = sh_h[laneM][k - 383];
        a1.x = sh_h[16 + laneM][k - 384];     a1.y = sh_h[16 + laneM][k - 383];
      }
      v2f b0;
      b0.x = Wt[(size_t)k * 1024 + nc];
      b0.y = Wt[(size_t)(k + 1) * 1024 + nc];
      acc0 = wmma4(a0, b0, acc0);
      acc1 = wmma4(a1, b0, acc1);
    }

    for (int j = 0; j < 8; ++j) {
      sh_g[j + 8 * laneG][lcol]      = acc0[j];
      sh_g[16 + j + 8 * laneG][lcol] = acc1[j];
    }
    __syncthreads();

    // pointwise LSTM cell for this hidden slice
    float* hdst = HG + ((size_t)(dir * 2 + (t & 1))) * B_SZ * HW2;
    for (int idx = threadIdx.x; idx < 32 * 32; idx += 256) {
      int m = idx >> 5, jj = idx & 31;
      int jg = j0 + jj;
      float gi = sh_g[m][jj]       + bias[jg];
      float gf = sh_g[m][32 + jj]  + bias[256 + jg];
      float gg = sh_g[m][64 + jj]  + bias[512 + jg];
      float go = sh_g[m][96 + jj]  + bias[768 + jg];
      float c  = sigm(gf) * sh_c[m][jj] + sigm(gi) * tanhf(gg);
      sh_c[m][jj] = c;
      float h = sigm(go) * tanhf(c);
      hdst[m * HW2 + jg] = h;
      OUT[((size_t)tw * B_SZ + m) * 512 + dir * 256 + jg] = h;
    }

    if (t < S_LEN - 1) {
      // device barrier across the NSLICE WGs of this direction
      __threadfence();           // make h stores device-visible (all threads)
      __syncthreads();
      if (threadIdx.x == 0) {
        unsigned* c = CNT + dir * S_LEN + t;
        atomicAdd(c, 1u);
        while (__hip_atomic_load(c, __ATOMIC_RELAXED, __HIP_MEMORY_SCOPE_AGENT)
               < (unsigned)NSLICE)
          __builtin_amdgcn_s_sleep(1);
        __threadfence();
      }
      __syncthreads();
    } else {
      __syncthreads();
    }
  }
}

// ---------------------------------------------------------------------------
// Emission: em[s][b][tag] = OUT[s][b][:] . emit_W[tag][:] + emit_b[tag]
// ---------------------------------------------------------------------------

__global__ void k_emit(const float* __restrict__ OUT, const float* __restrict__ eW,
                       const float* __restrict__ eb, float* __restrict__ EM) {
  int e = blockIdx.x * blockDim.x + threadIdx.x;
  if (e >= S_LEN * B_SZ * NTAG) return;
  int tag = e % NTAG;
  int sb = e / NTAG;
  const float* o = OUT + (size_t)sb * 512;
  const float* w = eW + (size_t)tag * 512;
  float s = eb[tag];
  for (int k = 0; k < 512; ++k) s += o[k] * w[k];
  EM[e] = s;
}

// ---------------------------------------------------------------------------
// CRF negative log-likelihood (sum over batch), single workgroup.
// ---------------------------------------------------------------------------

__global__ void __launch_bounds__(256) k_crf(
    const float* __restrict__ EM, const int* __restrict__ tags,
    const int* __restrict__ sent, const float* __restrict__ trans,
    const float* __restrict__ startv, const float* __restrict__ endv,
    float* __restrict__ out) {
  __shared__ float s_trans[NTAG][NTAG];
  __shared__ float s_alpha[B_SZ][NTAG + 1];
  __shared__ float s_next[B_SZ][NTAG + 1];
  __shared__ float s_num[B_SZ];
  __shared__ float s_den[B_SZ];
  __shared__ float s_start[NTAG], s_end[NTAG];

  for (int i = threadIdx.x; i < NTAG * NTAG; i += 256)
    s_trans[i / NTAG][i % NTAG] = trans[i];
  if (threadIdx.x < NTAG) {
    s_start[threadIdx.x] = startv[threadIdx.x];
    s_end[threadIdx.x] = endv[threadIdx.x];
  }
  __syncthreads();

  for (int i = threadIdx.x; i < B_SZ * NTAG; i += 256) {
    int b = i / NTAG, j = i % NTAG;
    s_alpha[b][j] = s_start[j] + EM[(0 * B_SZ + b) * NTAG + j];
  }

  if (threadIdx.x < B_SZ) {
    int b = threadIdx.x;
    int prev = tags[0 * B_SZ + b];
    float sc = s_start[prev] + EM[(0 * B_SZ + b) * NTAG + prev];
    for (int t = 1; t < S_LEN; ++t) {
      int tt = tags[t * B_SZ + b];
      float m = (sent[t * B_SZ + b] != 1) ? 1.0f : 0.0f;
      sc += (s_trans[prev][tt] + EM[(t * B_SZ + b) * NTAG + tt]) * m;
      if (m > 0.0f) prev = tt;
    }
    s_num[b] = sc + s_end[prev];
  }
  __syncthreads();

  for (int t = 1; t < S_LEN; ++t) {
    for (int i = threadIdx.x; i < B_SZ * NTAG; i += 256) {
      int b = i / NTAG, j = i % NTAG;
      float mx = -3.4e38f;
      for (int p = 0; p < NTAG; ++p)
        mx = fmaxf(mx, s_alpha[b][p] + s_trans[p][j]);
      float ss = 0.0f;
      for (int p = 0; p < NTAG; ++p)
        ss += __expf(s_alpha[b][p] + s_trans[p][j] - mx);
      float nxt = mx + __logf(ss) + EM[(t * B_SZ + b) * NTAG + j];
      float m = (sent[t * B_SZ + b] != 1) ? 1.0f : 0.0f;
      s_next[b][j] = (m > 0.0f) ? nxt : s_alpha[b][j];
    }
    __syncthreads();
    for (int i = threadIdx.x; i < B_SZ * NTAG; i += 256)
      s_alpha[i / NTAG][i % NTAG] = s_next[i / NTAG][i % NTAG];
    __syncthreads();
  }

  if (threadIdx.x < B_SZ) {
    int b = threadIdx.x;
    float mx = -3.4e38f;
    for (int j = 0; j < NTAG; ++j) mx = fmaxf(mx, s_alpha[b][j] + s_end[j]);
    float ss = 0.0f;
    for (int j = 0; j < NTAG; ++j) ss += __expf(s_alpha[b][j] + s_end[j] - mx);
    s_den[b] = mx + __logf(ss);
  }
  __syncthreads();

  if (threadIdx.x == 0) {
    float tot = 0.0f;
    for (int b = 0; b < B_SZ; ++b) tot += s_num[b] - s_den[b];
    out[0] = -tot;
  }
}

// ---------------------------------------------------------------------------
// Launch
// ---------------------------------------------------------------------------

extern "C" void kernel_launch(void* const* d_in, const int* in_sizes, int n_in,
                              void* d_out, int out_size, void* d_ws, size_t ws_size,
                              hipStream_t stream) {
  (void)in_sizes; (void)n_in; (void)out_size; (void)ws_size;
  const int*   sent   = (const int*)d_in[0];
  const int*   ch     = (const int*)d_in[1];
  const int*   tags   = (const int*)d_in[2];
  const float* Wwe    = (const float*)d_in[3];
  const float* Wce    = (const float*)d_in[4];
  const float* cWih_f = (const float*)d_in[5];
  const float* cWhh_f = (const float*)d_in[6];
  const float* cb_f   = (const float*)d_in[7];
  const float* wWih_f = (const float*)d_in[8];
  const float* wWhh_f = (const float*)d_in[9];
  const float* wb_f   = (const float*)d_in[10];
  const float* cWih_b = (const float*)d_in[11];
  const float* cWhh_b = (const float*)d_in[12];
  const float* cb_b   = (const float*)d_in[13];
  const float* wWih_b = (const float*)d_in[14];
  const float* wWhh_b = (const float*)d_in[15];
  const float* wb_b   = (const float*)d_in[16];
  const float* eW     = (const float*)d_in[17];
  const float* ebv    = (const float*)d_in[18];
  const float* trans  = (const float*)d_in[19];
  const float* startv = (const float*)d_in[20];
  const float* endv   = (const float*)d_in[21];

  float* ws  = (float*)d_ws;
  float* CE  = ws;                              // 16*4096*64
  float* WcT = CE + (size_t)CLEN * BS * DC;     // 2*128*256
  float* WwT = WcT + (size_t)2 * 128 * 256;     // 2*640*1024
  float* XW  = WwT + (size_t)2 * 640 * 1024;    // 4096*384
  float* OUT = XW + (size_t)BS * 384;           // 4096*512
  float* EM  = OUT + (size_t)BS * 512;          // 4096*17
  float* HG  = EM + (size_t)BS * NTAG;          // 2*2*32*256 hidden dbl-buffer
  unsigned* CNT = (unsigned*)(HG + (size_t)2 * 2 * B_SZ * HW2);  // 256 counters

  k_init<<<(2 * 2 * B_SZ * HW2 + 255) / 256, 256, 0, stream>>>(HG, CNT);
  k_char_gather<<<(CLEN * BS * DC + 255) / 256, 256, 0, stream>>>(ch, Wce, CE);
  k_word_gather<<<(S_LEN * B_SZ * DW + 255) / 256, 256, 0, stream>>>(sent, Wwe, XW);
  k_prep_wc<<<(2 * 128 * 256 + 255) / 256, 256, 0, stream>>>(cWih_f, cWhh_f, cWih_b, cWhh_b, WcT);
  k_prep_ww<<<(2 * 640 * 1024 + 255) / 256, 256, 0, stream>>>(wWih_f, wWhh_f, wWih_b, wWhh_b, WwT);

  dim3 cg(BS / 32, 2);
  k_char_lstm<<<cg, 256, 0, stream>>>(CE, WcT, cb_f, cb_b, XW);

  dim3 wg(NSLICE, 2);
  k_word_lstm<<<wg, 256, 0, stream>>>(XW, WwT, wb_f, wb_b, OUT, HG, CNT);

  k_emit<<<(S_LEN * B_SZ * NTAG + 255) / 256, 256, 0, stream>>>(OUT, eW, ebv, EM);
  k_crf<<<1, 256, 0, stream>>>(EM, tags, sent, trans, startv, endv, (float*)d_out);
}